// HypAgg_29532195127740
// MI455X (gfx1250) — compile-verified
//
#include <hip/hip_runtime.h>
#include <hip/hip_bf16.h>
#include <math.h>

typedef __attribute__((ext_vector_type(2))) float v2f;
typedef __attribute__((ext_vector_type(8))) float v8f;

#define N 768
#define D 128
#define MIN_NORM 1e-15f
#define PROJ_EPS 4e-3f
#define ART_CLIP (1.0f - 1e-7f)

__device__ __forceinline__ float artanh_clipped(float x) {
    // x >= 0 in all our uses; clip to 1-1e-7 like reference
    x = fminf(x, ART_CLIP);
    return 0.5f * __logf((1.0f + x) / (1.0f - x));
}

__device__ __forceinline__ float block_reduce_128(float v, float* sm) {
    int t = threadIdx.x;
    sm[t] = v;
    __syncthreads();
    #pragma unroll
    for (int s = 64; s > 0; s >>= 1) {
        if (t < s) sm[t] += sm[t + s];
        __syncthreads();
    }
    float r = sm[0];
    __syncthreads();
    return r;
}

// ---------------------------------------------------------------------------
// Kernel A: per-row stats. b_i = |x_i|^2, s_left_i, s_right_i, zero a[].
// grid = N blocks of 128 threads.
// ---------------------------------------------------------------------------
__global__ void row_stats_kernel(const float* __restrict__ x,
                                 const float* __restrict__ att_w,
                                 float* __restrict__ bvec,
                                 float* __restrict__ sl,
                                 float* __restrict__ sr,
                                 float* __restrict__ avec) {
    __shared__ float sm[128];
    int i = blockIdx.x;
    int d = threadIdx.x;
    float xv = x[i * D + d];
    float b  = block_reduce_128(xv * xv, sm);
    float dl = block_reduce_128(xv * att_w[d], sm);
    float dr = block_reduce_128(xv * att_w[D + d], sm);
    if (d == 0) {
        float pn = fmaxf(sqrtf(b), MIN_NORM);
        float t  = artanh_clipped(pn) / pn;   // x_tangent_i = t * x_i
        bvec[i] = b;
        sl[i]   = t * dl;
        sr[i]   = t * dr;
        avec[i] = 0.0f;
    }
}

// ---------------------------------------------------------------------------
// Kernel B: Gram tile via WMMA f32 16x16x4, then per-pair scalars.
// One wave (32 threads) per 16x16 tile. grid = (48, 48).
// Writes Mmat[i][j] = S_ij * beta_ij, atomically accumulates a[i] += S*alpha.
// ---------------------------------------------------------------------------
__global__ void gram_pair_kernel(const float* __restrict__ x,
                                 const float* __restrict__ adj,
                                 const float* __restrict__ bvec,
                                 const float* __restrict__ sl,
                                 const float* __restrict__ sr,
                                 const float* __restrict__ att_b,
                                 float* __restrict__ Mmat,
                                 float* __restrict__ avec) {
    const int j0   = blockIdx.x * 16;
    const int i0   = blockIdx.y * 16;
    const int lane = threadIdx.x;      // 0..31
    const int half = lane >> 4;        // 0/1 -> K group
    const int l16  = lane & 15;        // M (A) / N (B) within tile

    v8f acc = {};
    const float* arow = x + (i0 + l16) * D;  // A: row i0+l16
    const float* brow = x + (j0 + l16) * D;  // B: col j0+l16 of x^T = row of x
    #pragma unroll 4
    for (int k0 = 0; k0 < D; k0 += 4) {
        v2f a = *(const v2f*)(arow + k0 + 2 * half);
        v2f b = *(const v2f*)(brow + k0 + 2 * half);
        acc = __builtin_amdgcn_wmma_f32_16x16x4_f32(
            false, a, false, b, (short)0, acc, false, false);
    }

    const float bj   = bvec[j0 + l16];
    const float srj  = sr[j0 + l16];
    const float attb = att_b[0];

    float asum[8];
    #pragma unroll
    for (int v = 0; v < 8; ++v) {
        int row   = i0 + half * 8 + v;
        float g   = acc[v];
        float b_i = bvec[row];
        float sli = sl[row];

        float den   = fmaxf(1.0f - 2.0f * g + b_i * bj, MIN_NORM);
        float alpha = -(1.0f - 2.0f * g + bj) / den;
        float beta  = (1.0f - b_i) / den;

        float sn2 = alpha * alpha * b_i + 2.0f * alpha * beta * g + beta * beta * bj;
        float sn  = fmaxf(sqrtf(fmaxf(sn2, 0.0f)), MIN_NORM);
        float lam_fac = fmaxf(1.0f - b_i, MIN_NORM);      // 2/(sqrt_c*lambda_i)
        float w   = lam_fac * artanh_clipped(sn) / sn;

        float z   = sli + srj + attb;
        float att = (1.0f / (1.0f + __expf(-z))) * adj[row * N + j0 + l16];
        float S   = att * w;

        Mmat[row * N + j0 + l16] = S * beta;
        asum[v] = S * alpha;
    }

    // reduce asum over the 16 lanes of each half-wave, atomicAdd into a[row]
    #pragma unroll
    for (int v = 0; v < 8; ++v) {
        float val = asum[v];
        val += __shfl_xor(val, 1, 16);
        val += __shfl_xor(val, 2, 16);
        val += __shfl_xor(val, 4, 16);
        val += __shfl_xor(val, 8, 16);
        if (l16 == 0) atomicAdd(&avec[i0 + half * 8 + v], val);
    }
}

// ---------------------------------------------------------------------------
// Kernel C: support = M @ x via WMMA, + a_i * x_i fused at store.
// One wave per 16x16 output tile. grid = (8, 48).
// ---------------------------------------------------------------------------
__global__ void support_kernel(const float* __restrict__ Mmat,
                               const float* __restrict__ x,
                               const float* __restrict__ avec,
                               float* __restrict__ support) {
    const int j0   = blockIdx.x * 16;
    const int i0   = blockIdx.y * 16;
    const int lane = threadIdx.x;
    const int half = lane >> 4;
    const int l16  = lane & 15;

    v8f acc = {};
    const float* arow = Mmat + (i0 + l16) * N;
    #pragma unroll 4
    for (int k0 = 0; k0 < N; k0 += 4) {
        v2f a = *(const v2f*)(arow + k0 + 2 * half);
        v2f b;
        b.x = x[(k0 + 2 * half)     * D + j0 + l16];
        b.y = x[(k0 + 2 * half + 1) * D + j0 + l16];
        acc = __builtin_amdgcn_wmma_f32_16x16x4_f32(
            false, a, false, b, (short)0, acc, false, false);
    }

    #pragma unroll
    for (int v = 0; v < 8; ++v) {
        int row = i0 + half * 8 + v;
        int col = j0 + l16;
        support[row * D + col] = acc[v] + avec[row] * x[row * D + col];
    }
}

// ---------------------------------------------------------------------------
// Kernel D: output = proj(expmap(support_t, x)). Block-per-row, 128 threads.
// ---------------------------------------------------------------------------
__global__ void expmap_proj_kernel(const float* __restrict__ support,
                                   const float* __restrict__ x,
                                   const float* __restrict__ bvec,
                                   float* __restrict__ out) {
    __shared__ float sm[128];
    int i = blockIdx.x;
    int d = threadIdx.x;

    float u = support[i * D + d];
    float p = x[i * D + d];

    float u2 = block_reduce_128(u * u, sm);
    float pu = block_reduce_128(p * u, sm);

    float b_i   = bvec[i];
    float unorm = fmaxf(sqrtf(u2), MIN_NORM);
    float lam   = 2.0f / fmaxf(1.0f - b_i, MIN_NORM);
    float s     = tanhf(0.5f * lam * unorm) / unorm;   // second = s * u

    float y2 = s * s * u2;
    float xy = s * pu;
    float num_p = 1.0f + 2.0f * xy + y2;
    float num_s = (1.0f - b_i) * s;
    float den   = fmaxf(1.0f + 2.0f * xy + b_i * y2, MIN_NORM);
    float o     = (num_p * p + num_s * u) / den;

    float o2 = block_reduce_128(o * o, sm);
    float on = fmaxf(sqrtf(o2), MIN_NORM);
    float maxn = 1.0f - PROJ_EPS;     // sqrt(c)=1
    out[i * D + d] = (on > maxn) ? o * (maxn / on) : o;
}

// ---------------------------------------------------------------------------
extern "C" void kernel_launch(void* const* d_in, const int* in_sizes, int n_in,
                              void* d_out, int out_size, void* d_ws, size_t ws_size,
                              hipStream_t stream) {
    const float* x     = (const float*)d_in[0];   // [768,128]
    const float* adj   = (const float*)d_in[1];   // [768,768]
    const float* att_w = (const float*)d_in[2];   // [256,1]
    const float* att_b = (const float*)d_in[3];   // [1]
    float* out = (float*)d_out;

    char* ws = (char*)d_ws;
    float* Mmat    = (float*)(ws);                                   // 768*768
    float* support = (float*)(ws + (size_t)N * N * 4);               // 768*128
    float* bvec    = (float*)(ws + (size_t)N * N * 4 + (size_t)N * D * 4);
    float* sl      = bvec + N;
    float* sr      = sl + N;
    float* avec    = sr + N;

    row_stats_kernel<<<dim3(N), dim3(128), 0, stream>>>(x, att_w, bvec, sl, sr, avec);

    gram_pair_kernel<<<dim3(N / 16, N / 16), dim3(32), 0, stream>>>(
        x, adj, bvec, sl, sr, att_b, Mmat, avec);

    support_kernel<<<dim3(D / 16, N / 16), dim3(32), 0, stream>>>(
        Mmat, x, avec, support);

    expmap_proj_kernel<<<dim3(N), dim3(128), 0, stream>>>(support, x, bvec, out);
}